// DenseBackboneChunk_76338748719194
// MI455X (gfx1250) — compile-verified
//
#include <hip/hip_runtime.h>
#include <hip/hip_bf16.h>

#define HIDDEN 2048
#define INTER  5632
#define NLAYERS 6
#define KVW    256
#define EPSF   1e-6f

typedef float v2f __attribute__((ext_vector_type(2)));
typedef float v8f __attribute__((ext_vector_type(8)));

// ---------------------------------------------------------------- utilities
__global__ void k_copy(const float* __restrict__ src, float* __restrict__ dst, int n) {
    int i = blockIdx.x * blockDim.x + threadIdx.x;
    if (i < n) dst[i] = src[i];
}

// ---------------------------------------------------------------- RMSNorm: xn = h * rsqrt(mean(h^2)+eps) * w   (one block, 256 threads, n = 2048)
__global__ void k_rmsnorm(const float* __restrict__ h, const float* __restrict__ w,
                          float* __restrict__ xn) {
    __shared__ float red[256];
    int t = threadIdx.x;
    float v[8];
    float s = 0.f;
#pragma unroll
    for (int i = 0; i < 8; i++) { v[i] = h[t + i * 256]; s += v[i] * v[i]; }
    red[t] = s; __syncthreads();
    for (int off = 128; off > 0; off >>= 1) { if (t < off) red[t] += red[t + off]; __syncthreads(); }
    float scale = rsqrtf(red[0] / (float)HIDDEN + EPSF);
#pragma unroll
    for (int i = 0; i < 8; i++) xn[t + i * 256] = v[i] * scale * w[t + i * 256];
}

// ---------------------------------------------------------------- RMSNorm + scalar shared-gate: x2 = norm(h)*w ; g = sigmoid(dot(wshg, x2))
__global__ void k_rmsnorm_gate(const float* __restrict__ h, const float* __restrict__ w,
                               const float* __restrict__ wshg,
                               float* __restrict__ x2, float* __restrict__ gout) {
    __shared__ float red[256];
    int t = threadIdx.x;
    float v[8];
    float s = 0.f;
#pragma unroll
    for (int i = 0; i < 8; i++) { v[i] = h[t + i * 256]; s += v[i] * v[i]; }
    red[t] = s; __syncthreads();
    for (int off = 128; off > 0; off >>= 1) { if (t < off) red[t] += red[t + off]; __syncthreads(); }
    float scale = rsqrtf(red[0] / (float)HIDDEN + EPSF);
    float dot = 0.f;
#pragma unroll
    for (int i = 0; i < 8; i++) {
        float xv = v[i] * scale * w[t + i * 256];
        x2[t + i * 256] = xv;
        dot += wshg[t + i * 256] * xv;
    }
    __syncthreads();
    red[t] = dot; __syncthreads();
    for (int off = 128; off > 0; off >>= 1) { if (t < off) red[t] += red[t + off]; __syncthreads(); }
    if (t == 0) gout[0] = 1.f / (1.f + __expf(-red[0]));
}

// ---------------------------------------------------------------- attention scores partials: spart[b][w] = sum_{h in chunk b} q[h]*kvw[h][w]
// 16 blocks x 256 threads; kvw rows are read fully coalesced, q[h] is a uniform (scalar) load.
__global__ void k_scores_partial(const float* __restrict__ q, const float* __restrict__ kvw,
                                 float* __restrict__ spart) {
    int w = threadIdx.x;               // 0..255
    int b = blockIdx.x;                // 0..15 -> h chunk of 128
    const float* kp = kvw + (size_t)b * 128 * KVW + w;
    const float* qp = q + b * 128;
    float acc = 0.f;
#pragma unroll 4
    for (int hh = 0; hh < 128; hh++) acc += qp[hh] * kp[(size_t)hh * KVW];
    spart[b * KVW + w] = acc;
}

// ---------------------------------------------------------------- combine + softmax over the 256-wide window (one block)
__global__ void k_softmax(const float* __restrict__ spart, float* __restrict__ probs) {
    __shared__ float red[256];
    int t = threadIdx.x;
    float s = 0.f;
#pragma unroll
    for (int b = 0; b < 16; b++) s += spart[b * KVW + t];
    red[t] = s; __syncthreads();
    for (int off = 128; off > 0; off >>= 1) { if (t < off) red[t] = fmaxf(red[t], red[t + off]); __syncthreads(); }
    float m = red[0];
    __syncthreads();
    float e = __expf(s - m);
    red[t] = e; __syncthreads();
    for (int off = 128; off > 0; off >>= 1) { if (t < off) red[t] += red[t + off]; __syncthreads(); }
    probs[t] = e / red[0];
}

// ---------------------------------------------------------------- WMMA GEMV: partial[ck][m] = sum_{k in chunk ck} W[m][k]*x[k]
// Grid: x = 16-row tiles / 4 waves-per-block (exact), y = split-K chunk index.
// A = 16x4 f32 weight tile: lane L loads row L%16, cols k0+2*(L/16)..+1 (one
// non-temporal b64 per lane per step -- weights are stream-once, keep them out
// of L2 so kvw/activations stay resident). B = x segment replicated across all
// 16 columns (K = 2*half + vgpr), so every D column holds the same 16 dot
// products; V_WMMA_F32_16X16X4_F32 does the K=4 reduction + accumulate. Lanes
// 0 and 16 hold D column 0 (M=0..7 / M=8..15) and store the 16 results.
__global__ __launch_bounds__(128)
void k_gemv_wmma(const float* __restrict__ W, int ldw,
                 const float* __restrict__ x,
                 float* __restrict__ partial,
                 int M, int chunk) {
    int lane = threadIdx.x & 31;
    int tile = (blockIdx.x << 2) + (threadIdx.x >> 5);         // wave-uniform
    int ck   = blockIdx.y;
    int kbeg = ck * chunk;
    int half = (lane >> 4) << 1;                               // 0 or 2
    int row  = (tile << 4) + (lane & 15);

    const float* wp = W + (size_t)row * ldw + kbeg + half;
    const float* xp = x + kbeg + half;

    v8f c = {0.f, 0.f, 0.f, 0.f, 0.f, 0.f, 0.f, 0.f};
    int iters = chunk >> 2;
#pragma unroll 8
    for (int i = 0; i < iters; i++) {
        v2f a = __builtin_nontemporal_load((const v2f*)wp);    // streaming weights: TH=NT
        v2f b = *(const v2f*)xp;                               // hot x vector: regular temporal
        c = __builtin_amdgcn_wmma_f32_16x16x4_f32(false, a, false, b, (short)0, c, false, false);
        wp += 4; xp += 4;
    }
    if ((lane & 15) == 0) {            // lane 0 -> M=0..7, lane 16 -> M=8..15 (N=0 column)
        float* o = partial + (size_t)ck * M + (tile << 4) + ((lane >> 4) << 3);
#pragma unroll
        for (int i = 0; i < 8; i++) o[i] = c[i];
    }
}

// ---------------------------------------------------------------- deterministic split-K combines
__global__ void k_combine(const float* __restrict__ partial, int splitk, int M,
                          float* __restrict__ out) {
    int j = blockIdx.x * blockDim.x + threadIdx.x;
    if (j >= M) return;
    float s = 0.f;
    for (int i = 0; i < splitk; i++) s += partial[(size_t)i * M + j];
    out[j] = s;
}

__global__ void k_combine_add(const float* __restrict__ partial, int splitk, int M,
                              float* __restrict__ h) {
    int j = blockIdx.x * blockDim.x + threadIdx.x;
    if (j >= M) return;
    float s = 0.f;
    for (int i = 0; i < splitk; i++) s += partial[(size_t)i * M + j];
    h[j] += s;
}

__global__ void k_swiglu(const float* __restrict__ pg, const float* __restrict__ pu,
                         int splitk, float* __restrict__ t) {
    int j = blockIdx.x * blockDim.x + threadIdx.x;
    if (j >= INTER) return;
    float g = 0.f, u = 0.f;
    for (int i = 0; i < splitk; i++) {
        g += pg[(size_t)i * INTER + j];
        u += pu[(size_t)i * INTER + j];
    }
    float si = g / (1.f + __expf(-g));      // silu
    t[j] = si * u;
}

__global__ void k_combine_gated_add(const float* __restrict__ partial, int splitk, int M,
                                    const float* __restrict__ gscalar,
                                    float* __restrict__ h) {
    int j = blockIdx.x * blockDim.x + threadIdx.x;
    if (j >= M) return;
    float s = 0.f;
    for (int i = 0; i < splitk; i++) s += partial[(size_t)i * M + j];
    h[j] += gscalar[0] * s;
}

// ---------------------------------------------------------------- launch
extern "C" void kernel_launch(void* const* d_in, const int* in_sizes, int n_in,
                              void* d_out, int out_size, void* d_ws, size_t ws_size,
                              hipStream_t stream) {
    const float* x_in  = (const float*)d_in[0];   // [2048]
    const float* kvw   = (const float*)d_in[1];   // [2048][256]
    const float* n1w   = (const float*)d_in[2];   // [6][2048]
    const float* n2w   = (const float*)d_in[3];   // [6][2048]
    const float* qw    = (const float*)d_in[4];   // [6][2048][2048]
    // d_in[5] (k_w), d_in[6] (v_w), d_in[8] (router_w): outputs unused -> skipped (~205 MB saved)
    const float* ow    = (const float*)d_in[7];   // [6][2048][2048]
    const float* shgw  = (const float*)d_in[9];   // [6][1][2048]
    const float* gatew = (const float*)d_in[10];  // [6][5632][2048]
    const float* upw   = (const float*)d_in[11];  // [6][5632][2048]
    const float* downw = (const float*)d_in[12];  // [6][2048][5632]

    float* h = (float*)d_out;                     // hidden state lives in d_out (2048 f32)

    float* ws    = (float*)d_ws;
    float* xn    = ws;  ws += HIDDEN;
    float* q     = ws;  ws += HIDDEN;
    float* ctx   = ws;  ws += HIDDEN;
    float* x2    = ws;  ws += HIDDEN;
    float* gbuf  = ws;  ws += 16;
    float* spart = ws;  ws += 16 * KVW;
    float* probs = ws;  ws += KVW;
    float* t     = ws;  ws += INTER;
    float* pA    = ws;  ws += 16 * HIDDEN;        // partials for 2048-row GEMVs
    float* pg    = ws;  ws += 8 * INTER;          // gate partials
    float* pu    = ws;  ws += 8 * INTER;          // up partials
    // total ~564 KB of workspace

    k_copy<<<(HIDDEN + 255) / 256, 256, 0, stream>>>(x_in, h, HIDDEN);

    for (int l = 0; l < NLAYERS; l++) {
        const float* Wq = qw    + (size_t)l * HIDDEN * HIDDEN;
        const float* Wo = ow    + (size_t)l * HIDDEN * HIDDEN;
        const float* Wg = gatew + (size_t)l * INTER * HIDDEN;
        const float* Wu = upw   + (size_t)l * INTER * HIDDEN;
        const float* Wd = downw + (size_t)l * HIDDEN * INTER;

        // xn = rmsnorm(h) * n1
        k_rmsnorm<<<1, 256, 0, stream>>>(h, n1w + (size_t)l * HIDDEN, xn);

        // q = Wq @ xn   (tiles = 128, split-K = 16 -> 2048 waves)
        {
            const int sk = 16;
            k_gemv_wmma<<<dim3(HIDDEN / 16 / 4, sk), 128, 0, stream>>>(
                Wq, HIDDEN, xn, pA, HIDDEN, HIDDEN / sk);
            k_combine<<<HIDDEN / 256, 256, 0, stream>>>(pA, sk, HIDDEN, q);
        }

        // attention over kv window: probs = softmax(q @ kvw); ctx = kvw @ probs
        k_scores_partial<<<16, 256, 0, stream>>>(q, kvw, spart);
        k_softmax<<<1, 256, 0, stream>>>(spart, probs);
        {
            const int sk = 4;
            k_gemv_wmma<<<dim3(HIDDEN / 16 / 4, sk), 128, 0, stream>>>(
                kvw, KVW, probs, pA, HIDDEN, KVW / sk);
            k_combine<<<HIDDEN / 256, 256, 0, stream>>>(pA, sk, HIDDEN, ctx);
        }

        // h += Wo @ ctx
        {
            const int sk = 16;
            k_gemv_wmma<<<dim3(HIDDEN / 16 / 4, sk), 128, 0, stream>>>(
                Wo, HIDDEN, ctx, pA, HIDDEN, HIDDEN / sk);
            k_combine_add<<<HIDDEN / 256, 256, 0, stream>>>(pA, sk, HIDDEN, h);
        }

        // x2 = rmsnorm(h) * n2 ; g = sigmoid(Wshg @ x2)
        k_rmsnorm_gate<<<1, 256, 0, stream>>>(h, n2w + (size_t)l * HIDDEN,
                                              shgw + (size_t)l * HIDDEN, x2, gbuf);

        // SwiGLU: t = silu(Wg@x2) * (Wu@x2)   (tiles = 352, split-K = 8 -> 2816 waves each)
        {
            const int sk = 8;
            k_gemv_wmma<<<dim3(INTER / 16 / 4, sk), 128, 0, stream>>>(
                Wg, HIDDEN, x2, pg, INTER, HIDDEN / sk);
            k_gemv_wmma<<<dim3(INTER / 16 / 4, sk), 128, 0, stream>>>(
                Wu, HIDDEN, x2, pu, INTER, HIDDEN / sk);
            k_swiglu<<<(INTER + 255) / 256, 256, 0, stream>>>(pg, pu, sk, t);
        }

        // h += g * (Wd @ t)
        {
            const int sk = 16;   // K = 5632 -> 352 per wave (88 WMMA steps)
            k_gemv_wmma<<<dim3(HIDDEN / 16 / 4, sk), 128, 0, stream>>>(
                Wd, INTER, t, pA, HIDDEN, INTER / sk);
            k_combine_gated_add<<<HIDDEN / 256, 256, 0, stream>>>(pA, sk, HIDDEN, gbuf, h);
        }
    }
}